// entropy_loss_73358041415727
// MI455X (gfx1250) — compile-verified
//
#include <hip/hip_runtime.h>
#include <hip/hip_bf16.h>

#define NBINS   256
#define JBINS   (NBINS * NBINS)          // 65536 bins, 256 KB as u32
#define OUT_H   112
#define OUT_W   96
#define BC      768                       // 256 batch * 3 channels
#define N_ELEM  (BC * OUT_H * OUT_W)      // 8,257,536
#define X0_H    224
#define X0_W    192
#define EPSC    1e-8f

typedef float v2f __attribute__((ext_vector_type(2)));
typedef float v4f __attribute__((ext_vector_type(4)));
typedef float v8f __attribute__((ext_vector_type(8)));

// ---------------------------------------------------------------------------
// Kernel 0: zero the global joint histogram in workspace
// ---------------------------------------------------------------------------
__global__ void zero_hist(unsigned int* __restrict__ gjoint) {
    int i = blockIdx.x * blockDim.x + threadIdx.x;
    if (i < JBINS) gjoint[i] = 0u;
}

// ---------------------------------------------------------------------------
// Kernel 1: joint histogram, privatized in 256 KB of LDS per workgroup
// (CDNA5: 320 KB LDS/WGP makes the FULL 256x256 joint histogram fit in LDS).
// Nearest resize of x0 (224x192 -> 112x96) is exactly a stride-2 gather.
// Each thread processes 4 output pixels per iteration via 128-bit loads.
// ---------------------------------------------------------------------------
__global__ __launch_bounds__(1024) void joint_hist(
        const float* __restrict__ x0,
        const float* __restrict__ img,
        unsigned int* __restrict__ gjoint) {
    extern __shared__ unsigned int lhist[];            // JBINS entries (256 KB)

    for (int i = threadIdx.x; i < JBINS; i += blockDim.x) lhist[i] = 0u;
    __syncthreads();

    const int NV   = N_ELEM / 4;                       // 2,064,384 float4 groups
    const int tid  = blockIdx.x * blockDim.x + threadIdx.x;
    const int nthr = gridDim.x * blockDim.x;

    for (int v = tid; v < NV; v += nthr) {
        const int j4 = v % (OUT_W / 4);                // 0..23  (4 output cols)
        const int t  = v / (OUT_W / 4);
        const int i  = t % OUT_H;                      // 0..111 output row
        const int bc = t / OUT_H;                      // 0..767 batch*channel

        // img: contiguous 128-bit load (streamed once -> non-temporal)
        const v4f iv = __builtin_nontemporal_load(
            (const v4f*)(img + ((size_t)(bc * OUT_H + i) * OUT_W + 4 * j4)));
        // x0: row 2*i, cols 8*j4 .. 8*j4+7 -> use elements 0 and 2 of each v4f
        const float* xrow = x0 + ((size_t)bc * X0_H + 2 * i) * X0_W + 8 * j4;
        const v4f xa = __builtin_nontemporal_load((const v4f*)(xrow));
        const v4f xb = __builtin_nontemporal_load((const v4f*)(xrow + 4));

        // uint8 truncation (values in [0,255))
        const unsigned ix0 = (unsigned)iv.x, ix1 = (unsigned)iv.y;
        const unsigned ix2 = (unsigned)iv.z, ix3 = (unsigned)iv.w;
        const unsigned iy0 = (unsigned)xa.x, iy1 = (unsigned)xa.z;
        const unsigned iy2 = (unsigned)xb.x, iy3 = (unsigned)xb.z;

        atomicAdd(&lhist[ix0 * NBINS + iy0], 1u);      // ds_add_u32
        atomicAdd(&lhist[ix1 * NBINS + iy1], 1u);
        atomicAdd(&lhist[ix2 * NBINS + iy2], 1u);
        atomicAdd(&lhist[ix3 * NBINS + iy3], 1u);
    }
    __syncthreads();

    // one flush per block; skip empty bins
    for (int i = threadIdx.x; i < JBINS; i += blockDim.x) {
        const unsigned c = lhist[i];
        if (c) atomicAdd(&gjoint[i], c);               // global_atomic_add_u32
    }
}

// ---------------------------------------------------------------------------
// Kernel 2: marginals via V_WMMA_F32_16X16X4_F32 (exact: counts < 2^24),
// then entropy sums. Single block, 256 threads (8 waves).
//   hist_x = J * ones   (row sums),  hist_y = J^T * ones (via transposed gather)
// A-layout (16x4 f32): lane L -> M = L&15, holds K = 2*(L>>4) + {0,1}.
// C/D layout: VGPR g, lane L -> M = g + 8*(L>>4), N = L&15 (all N identical
// here since B is all-ones, so any N column carries the row sum).
// ---------------------------------------------------------------------------
__global__ __launch_bounds__(256) void entropy_reduce(
        const unsigned int* __restrict__ gjoint,
        float* __restrict__ out) {
    __shared__ float histx[NBINS];
    __shared__ float histy[NBINS];
    __shared__ float red[256];

    const int t    = threadIdx.x;
    const int lane = t & 31;
    const int wave = t >> 5;

    v2f bones; bones.x = 1.0f; bones.y = 1.0f;         // B = ones (layout-free)

    // 32 WMMA chains: 16 slabs of hist_x + 16 slabs of hist_y; 4 per wave.
    const int M     = lane & 15;
    const int kbase = (lane >> 4) * 2;
    for (int c = wave * 4; c < wave * 4 + 4; ++c) {
        const bool is_y = (c >= 16);
        const int  slab = is_y ? (c - 16) : c;
        const int  row  = slab * 16 + M;
        v8f acc = {};
        for (int k4 = 0; k4 < 64; ++k4) {              // K = 256 in steps of 4
            const int K0 = k4 * 4 + kbase;
            v2f a;
            if (!is_y) {                               // A[M][K] = J[row][K]
                a.x = (float)gjoint[row * NBINS + K0];
                a.y = (float)gjoint[row * NBINS + K0 + 1];
            } else {                                   // A[M][K] = J[K][row]
                a.x = (float)gjoint[K0 * NBINS + row];
                a.y = (float)gjoint[(K0 + 1) * NBINS + row];
            }
            acc = __builtin_amdgcn_wmma_f32_16x16x4_f32(
                false, a, false, bones, (short)0, acc, false, false);
        }
        if ((lane & 15) == 0) {                        // N==0 column holds sums
            float* dst = is_y ? histy : histx;
            const int mbase = slab * 16 + 8 * (lane >> 4);
            #pragma unroll
            for (int g = 0; g < 8; ++g) dst[mbase + g] = acc[g];
        }
    }
    __syncthreads();

    // entropy: result = -sum px*log(px+e) - sum py*log(py+e) + sum pxy*log(pxy+e)
    const float inv = 1.0f / 96.0f;                    // faithful quirk: size=96
    float part = 0.0f;
    {
        const float px = histx[t] * inv;
        const float py = histy[t] * inv;
        part -= px * logf(px + EPSC);
        part -= py * logf(py + EPSC);
    }
    for (int j = 0; j < NBINS; ++j) {
        const float pxy = (float)gjoint[t * NBINS + j] * inv;
        part += pxy * logf(pxy + EPSC);
    }
    red[t] = part;
    __syncthreads();
    for (int s = 128; s > 0; s >>= 1) {
        if (t < s) red[t] += red[t + s];
        __syncthreads();
    }
    if (t == 0) out[0] = red[0];                       // LOSS_WEIGHT = 1
}

// ---------------------------------------------------------------------------
extern "C" void kernel_launch(void* const* d_in, const int* in_sizes, int n_in,
                              void* d_out, int out_size, void* d_ws, size_t ws_size,
                              hipStream_t stream) {
    const float* x0  = (const float*)d_in[0];          // (256,3,224,192)
    const float* img = (const float*)d_in[1];          // (256,3,112,96)
    unsigned int* gjoint = (unsigned int*)d_ws;        // 65536 u32 = 256 KB
    float* out = (float*)d_out;

    zero_hist<<<JBINS / 256, 256, 0, stream>>>(gjoint);
    joint_hist<<<64, 1024, JBINS * sizeof(unsigned int), stream>>>(x0, img, gjoint);
    entropy_reduce<<<1, 256, 0, stream>>>(gjoint, out);
}